// DeepFit_18794776887995
// MI455X (gfx1250) — compile-verified
//
#include <hip/hip_runtime.h>
#include <math.h>

typedef __attribute__((ext_vector_type(2))) float v2f;
typedef __attribute__((ext_vector_type(8))) float v8f;

#define NPTS 256
#define KNN  20
#define ROWP 258   // even pad: row base 8B-aligned for ds_load_b64; bank=(2i+j)%64 conflict-free
#define NEG_INF (-3.4e38f)

__device__ inline float wave_sum(float v) {
#pragma unroll
  for (int off = 16; off > 0; off >>= 1) v += __shfl_xor(v, off, 32);
  return v;
}

__global__ __launch_bounds__(256) void deepfit_kernel(
    const float* __restrict__ points, const float* __restrict__ weights,
    float* __restrict__ out_feat, float* __restrict__ out_beta,
    float* __restrict__ out_nest, float* __restrict__ out_nn)
{
  extern __shared__ float smem[];
  float* dist = smem;                          // 256*258 floats
  v2f*   ab   = (v2f*)(dist + NPTS * ROWP);    // [2][256] packed WMMA A/B fragments
  float* px   = (float*)(ab + 2 * NPTS);       // 256
  float* py   = px + NPTS;
  float* pz   = py + NPTS;
  float* xx   = pz + NPTS;                     // 256
  int*   nbr  = (int*)(xx + NPTS);             // 256*20
  float* red  = (float*)(nbr + NPTS * KNN);    // 8 waves * 32 slots
  float* scal = red + 8 * 32;                  // [0]=h [1]=use_w [2..7]=beta

  const int bat  = blockIdx.x;
  const int tid  = threadIdx.x;
  const int lane = tid & 31;
  const int wid  = tid >> 5;

  // prefetch weights row (consumed only after the KNN phase)
  __builtin_prefetch(weights + (size_t)bat * NPTS + tid, 0, 3);

  // ---- stage points into LDS ----
  const float* pbase = points + (size_t)bat * 3 * NPTS;
  const float xi = pbase[tid];
  const float yi = pbase[NPTS + tid];
  const float zi = pbase[2 * NPTS + tid];
  px[tid] = xi; py[tid] = yi; pz[tid] = zi;
  xx[tid] = xi * xi + yi * yi + zi * zi;
  ab[tid]        = (v2f){ xi, yi };   // K=0,1 fragment (lanes 0-15)
  ab[NPTS + tid] = (v2f){ zi, 0.0f }; // K=2,3 fragment (lanes 16-31), K=3 zero-pad
  __syncthreads();

  // ---- Gram matrix G = X^T X via V_WMMA_F32_16X16X4_F32 (K=3 padded to 4) ----
  // pdist[i][j] = 2*G[i][j] - xx[i] - xx[j]
  {
    const int m     = lane & 15;
    const int abase = (lane >= 16) ? NPTS : 0;  // uniform address select, no branch
    const int hi8   = (lane >= 16) ? 8 : 0;
#pragma unroll 2
    for (int t = wid; t < 256; t += 8) {
      const int ti = t >> 4, tj = t & 15;
      const v2f av = ab[abase + ti * 16 + m];
      const v2f bv = ab[abase + tj * 16 + m];
      v8f acc = {};
      acc = __builtin_amdgcn_wmma_f32_16x16x4_f32(
          false, av, false, bv, (short)0, acc, false, false);
#pragma unroll
      for (int r = 0; r < 8; ++r) {
        const int gi = ti * 16 + r + hi8;
        const int gj = tj * 16 + m;
        dist[gi * ROWP + gj] = 2.0f * acc[r] - xx[gi] - xx[gj];
      }
    }
  }
  __syncthreads();

  // ---- top-K (largest pdist = nearest); strict > keeps lowest index on ties ----
  {
    const v2f* rowv = (const v2f*)(dist + tid * ROWP);  // 8B-aligned (ROWP even)
    float* row = dist + tid * ROWP;
    for (int kk = 0; kk < KNN; ++kk) {
      float best = NEG_INF; int bj = 0;
      for (int j2 = 0; j2 < NPTS / 2; ++j2) {
        const v2f v = rowv[j2];
        if (v.x > best) { best = v.x; bj = 2 * j2; }
        if (v.y > best) { best = v.y; bj = 2 * j2 + 1; }
      }
      row[bj] = NEG_INF;
      nbr[tid * KNN + kk] = bj;
    }
  }
  __syncthreads();

  // ---- feature tensor (B,10,N,K): [d2, dx,dy,dz, cx,cy,cz, nx,ny,nz] ----
  {
    float dx[KNN], dy[KNN], dz[KNN];
#pragma unroll
    for (int kk = 0; kk < KNN; ++kk) {
      const int j = nbr[tid * KNN + kk];
      dx[kk] = px[j] - xi; dy[kk] = py[j] - yi; dz[kk] = pz[j] - zi;
    }
    const size_t CS = (size_t)NPTS * KNN;  // 5120
    float* fb = out_feat + (size_t)bat * 10 * CS + (size_t)tid * KNN;
#pragma unroll
    for (int q = 0; q < 5; ++q) {
      const int k0 = 4 * q;
      float4 t;
      t.x = dx[k0+0]*dx[k0+0] + dy[k0+0]*dy[k0+0] + dz[k0+0]*dz[k0+0];
      t.y = dx[k0+1]*dx[k0+1] + dy[k0+1]*dy[k0+1] + dz[k0+1]*dz[k0+1];
      t.z = dx[k0+2]*dx[k0+2] + dy[k0+2]*dy[k0+2] + dz[k0+2]*dz[k0+2];
      t.w = dx[k0+3]*dx[k0+3] + dy[k0+3]*dy[k0+3] + dz[k0+3]*dz[k0+3];
      *(float4*)(fb + 0 * CS + k0) = t;
      *(float4*)(fb + 1 * CS + k0) = make_float4(dx[k0], dx[k0+1], dx[k0+2], dx[k0+3]);
      *(float4*)(fb + 2 * CS + k0) = make_float4(dy[k0], dy[k0+1], dy[k0+2], dy[k0+3]);
      *(float4*)(fb + 3 * CS + k0) = make_float4(dz[k0], dz[k0+1], dz[k0+2], dz[k0+3]);
      *(float4*)(fb + 4 * CS + k0) = make_float4(xi, xi, xi, xi);
      *(float4*)(fb + 5 * CS + k0) = make_float4(yi, yi, yi, yi);
      *(float4*)(fb + 6 * CS + k0) = make_float4(zi, zi, zi, zi);
      *(float4*)(fb + 7 * CS + k0) = make_float4(xi+dx[k0], xi+dx[k0+1], xi+dx[k0+2], xi+dx[k0+3]);
      *(float4*)(fb + 8 * CS + k0) = make_float4(yi+dy[k0], yi+dy[k0+1], yi+dy[k0+2], yi+dy[k0+3]);
      *(float4*)(fb + 9 * CS + k0) = make_float4(zi+dz[k0], zi+dz[k0+1], zi+dz[k0+2], zi+dz[k0+3]);
    }
  }

  // ---- wjet fit: h, valid_count ----
  const float wgt = weights[(size_t)bat * NPTS + tid];
  {
    const float sax = wave_sum(fabsf(xi));
    const float say = wave_sum(fabsf(yi));
    const float sc  = wave_sum(wgt > 0.001f ? 1.0f : 0.0f);
    if (lane == 0) { red[wid*32+0] = sax; red[wid*32+1] = say; red[wid*32+2] = sc; }
  }
  __syncthreads();
  if (tid == 0) {
    float ax = 0.f, ay = 0.f, cc = 0.f;
    for (int q = 0; q < 8; ++q) { ax += red[q*32+0]; ay += red[q*32+1]; cc += red[q*32+2]; }
    float h = (ax / 256.0f + ay / 256.0f) * 0.5f;
    if (fabsf(h) < 1e-4f) h = 0.1f;
    scal[0] = h;
    scal[1] = (cc > 18.0f) ? 1.0f : 0.0f;
  }
  __syncthreads();

  const float h   = scal[0];
  const float wv2 = (scal[1] > 0.5f) ? wgt : 1.0f;
  const float xs = xi / h, ys = yi / h;
  const float a6[6] = { xs, ys, xs * xs, ys * ys, xs * ys, 1.0f };

  // ---- 27 moment reductions (21 XtX upper + 6 XtY) ----
  {
    float mom[27];
    int c = 0;
    for (int i = 0; i < 6; ++i)
      for (int j = i; j < 6; ++j) mom[c++] = a6[i] * a6[j] * wv2;
    for (int i = 0; i < 6; ++i) mom[21 + i] = a6[i] * zi * wv2;
#pragma unroll
    for (int q = 0; q < 27; ++q) {
      const float s = wave_sum(mom[q]);
      if (lane == 0) red[wid * 32 + q] = s;
    }
  }
  __syncthreads();

  // ---- 6x6 Cholesky solve + scaling + n_est (single thread) ----
  if (tid == 0) {
    float M[6][6], Y[6];
    {
      int c = 0;
      for (int i = 0; i < 6; ++i)
        for (int j = i; j < 6; ++j) {
          float s = 0.f;
          for (int q = 0; q < 8; ++q) s += red[q * 32 + c];
          M[i][j] = M[j][i] = s; ++c;
        }
      for (int i = 0; i < 6; ++i) {
        float s = 0.f;
        for (int q = 0; q < 8; ++q) s += red[q * 32 + 21 + i];
        Y[i] = s;
      }
    }
    float L[6][6];
    for (int j = 0; j < 6; ++j) {
      float d = M[j][j];
      for (int k = 0; k < j; ++k) d -= L[j][k] * L[j][k];
      L[j][j] = sqrtf(d);
      for (int i = j + 1; i < 6; ++i) {
        float s = M[i][j];
        for (int k = 0; k < j; ++k) s -= L[i][k] * L[j][k];
        L[i][j] = s / L[j][j];
      }
    }
    float tt[6], be[6];
    for (int i = 0; i < 6; ++i) {
      float s = Y[i];
      for (int k = 0; k < i; ++k) s -= L[i][k] * tt[k];
      tt[i] = s / L[i][i];
    }
    for (int i = 5; i >= 0; --i) {
      float s = tt[i];
      for (int k = i + 1; k < 6; ++k) s -= L[k][i] * be[k];
      be[i] = s / L[i][i];
    }
    const float h2 = h * h;
    const float dv[6] = { h, h, h2, h2, h2, 1.0f };
    for (int i = 0; i < 6; ++i) be[i] /= dv[i];
    const float nx = -be[0], ny = -be[1];
    float nrm = sqrtf(nx * nx + ny * ny + 1.0f);
    nrm = fmaxf(nrm, 1e-12f);
    out_nest[(size_t)bat * 3 + 0] = nx / nrm;
    out_nest[(size_t)bat * 3 + 1] = ny / nrm;
    out_nest[(size_t)bat * 3 + 2] = 1.0f / nrm;
    for (int i = 0; i < 6; ++i) {
      out_beta[(size_t)bat * 6 + i] = be[i];
      scal[2 + i] = be[i];
    }
  }
  __syncthreads();

  // ---- per-point neighbor normals ----
  {
    const float b0 = scal[2], b1 = scal[3], b2 = scal[4], b3 = scal[5], b4 = scal[6];
    const float nx = -(b0 + 2.0f * b2 * xs + b4 * ys);
    const float ny = -(b1 + 2.0f * b3 * ys + b4 * xs);
    float nrm = sqrtf(nx * nx + ny * ny + 1.0f);
    nrm = fmaxf(nrm, 1e-12f);
    float* onn = out_nn + ((size_t)bat * NPTS + tid) * 3;
    onn[0] = nx / nrm; onn[1] = ny / nrm; onn[2] = 1.0f / nrm;
  }
}

extern "C" void kernel_launch(void* const* d_in, const int* in_sizes, int n_in,
                              void* d_out, int out_size, void* d_ws, size_t ws_size,
                              hipStream_t stream) {
  const float* points  = (const float*)d_in[0];   // (B,3,256)
  const float* weights = (const float*)d_in[1];   // (B,256)
  float* out = (float*)d_out;

  const int B = in_sizes[0] / (3 * NPTS);         // 512
  const size_t CS     = (size_t)NPTS * KNN;
  const size_t F_BETA = (size_t)B * 10 * CS;      // after feature
  const size_t F_NEST = F_BETA + (size_t)B * 6;
  const size_t F_NN   = F_NEST + (size_t)B * 3;

  const size_t smem_bytes =
      (size_t)(NPTS * ROWP) * sizeof(float) +              // dist
      (size_t)(2 * NPTS) * sizeof(v2f) +                   // ab fragments
      (size_t)(4 * NPTS) * sizeof(float) +                 // px/py/pz/xx
      (size_t)(NPTS * KNN) * sizeof(int) +                 // nbr
      (size_t)(8 * 32 + 16) * sizeof(float);               // red + scal  (~294 KB)

  hipFuncSetAttribute((const void*)deepfit_kernel,
                      hipFuncAttributeMaxDynamicSharedMemorySize,
                      (int)smem_bytes);

  deepfit_kernel<<<B, 256, smem_bytes, stream>>>(
      points, weights, out, out + F_BETA, out + F_NEST, out + F_NN);
}